// CompGCNLayer_56118042690063
// MI455X (gfx1250) — compile-verified
//
#include <hip/hip_runtime.h>

#define N_NODES 50000
#define N_EDGES 625000
#define N_RELS  500
#define DIM     128
#define BN_EPS  1e-5f

typedef __attribute__((ext_vector_type(2))) float v2f;
typedef __attribute__((ext_vector_type(8))) float v8f;

// ---------------------------------------------------------------------------
// Kernel 1: zero the workspace (3 direction accumulators + column stats)
// ---------------------------------------------------------------------------
__global__ __launch_bounds__(256) void zero_ws4(float4* __restrict__ ws, long n4) {
    long i = (long)blockIdx.x * blockDim.x + threadIdx.x;
    long stride = (long)gridDim.x * blockDim.x;
    float4 z = make_float4(0.f, 0.f, 0.f, 0.f);
    for (; i < n4; i += stride) ws[i] = z;
}

// ---------------------------------------------------------------------------
// Kernel 2: edge scatter. One wave32 per edge; lane handles 4 of 128 dims.
// acc[dir][dst] += norm * (ent[src] - rel[type])
// ent_emb (25.6MB) + acc (76.8MB) are L2-resident -> gathers/atomics hit L2.
// ---------------------------------------------------------------------------
__global__ __launch_bounds__(256) void scatter_edges(
        const float* __restrict__ ent, const float* __restrict__ rel,
        const int* __restrict__ eidx, const int* __restrict__ etype,
        const int* __restrict__ edir, const float* __restrict__ enorm,
        float* __restrict__ acc) {
    const int lane = threadIdx.x & 31;
    const int e = blockIdx.x * (blockDim.x >> 5) + (threadIdx.x >> 5);
    if (e >= N_EDGES) return;
    const int   src = eidx[e];
    const int   dst = eidx[N_EDGES + e];
    const int   typ = etype[e];
    const int   dir = edir[e];
    const float w   = enorm[e];
    const float4 a = *((const float4*)(ent + (long)src * DIM) + lane);
    const float4 b = *((const float4*)(rel + (long)typ * DIM) + lane);
    float* o = acc + ((long)dir * N_NODES + dst) * DIM + lane * 4;
    atomicAdd(o + 0, w * (a.x - b.x));
    atomicAdd(o + 1, w * (a.y - b.y));
    atomicAdd(o + 2, w * (a.z - b.z));
    atomicAdd(o + 3, w * (a.w - b.w));
}

// ---------------------------------------------------------------------------
// WMMA helper: C += A(16x128 rows of `A`) x B(128x16, B[k][n] = W[n][k])
// using V_WMMA_F32_16X16X4_F32, K stepped by 4.
//   A frag: lanes 0-15 -> M=lane, K={0,1}; lanes 16-31 -> M=lane-16, K={2,3}
//   B frag: lanes 0-15 -> K={0,1}, N=lane; lanes 16-31 -> K={2,3}, N=lane-16
// Per-lane both fragments are contiguous float2 loads (row-major, stride DIM).
// ---------------------------------------------------------------------------
__device__ __forceinline__ v8f wmma_k128(v8f c, const float* __restrict__ Arow,
                                         const float* __restrict__ Brow) {
#pragma unroll 8
    for (int k = 0; k < DIM; k += 4) {
        v2f a = *(const v2f*)(Arow + k);
        v2f b = *(const v2f*)(Brow + k);
        c = __builtin_amdgcn_wmma_f32_16x16x4_f32(
                /*neg_a=*/false, a, /*neg_b=*/false, b,
                /*c_mod=*/(short)0, c, /*reuse_a=*/false, /*reuse_b=*/false);
    }
    return c;
}

// ---------------------------------------------------------------------------
// Kernel 3: out[m,n] = sum_d sum_k acc[d][m,k] * W_d[n,k]
// Block = 256 threads = 8 waves; block -> 16-node row tile, wave -> 16-col tile.
// ---------------------------------------------------------------------------
__global__ __launch_bounds__(256) void gemm_out(
        const float* __restrict__ acc, const float* __restrict__ Wout,
        const float* __restrict__ Win, const float* __restrict__ Wloop,
        float* __restrict__ out) {
    const int lane = threadIdx.x & 31;
    const int wave = threadIdx.x >> 5;       // 0..7 -> N tile
    const int m0 = blockIdx.x * 16;          // node row tile (50000/16 = 3125)
    const int n0 = wave * 16;
    const int ml = lane & 15;
    const int kh = (lane >> 4) * 2;          // K half-select: 0 or 2

    const long aoff = (long)(m0 + ml) * DIM + kh;
    const long boff = (long)(n0 + ml) * DIM + kh;

    v8f c = {};
    c = wmma_k128(c, acc + 0L * N_NODES * DIM + aoff, Wout  + boff);
    c = wmma_k128(c, acc + 1L * N_NODES * DIM + aoff, Win   + boff);
    c = wmma_k128(c, acc + 2L * N_NODES * DIM + aoff, Wloop + boff);

    // C/D layout: VGPR r -> row m0 + r + 8*(lane>=16), col n0 + (lane&15)
    float* o = out + (long)(m0 + (lane >> 4) * 8) * DIM + n0 + ml;
#pragma unroll
    for (int r = 0; r < 8; ++r) o[(long)r * DIM] = c[r];
}

// ---------------------------------------------------------------------------
// Kernel 4: per-column sum / sum-of-squares of out (pre-BN), via LDS + atomics
// ---------------------------------------------------------------------------
__global__ __launch_bounds__(256) void col_stats(
        const float* __restrict__ out, float* __restrict__ colsum,
        float* __restrict__ colsq) {
    __shared__ float s_sum[2][DIM];
    __shared__ float s_sq[2][DIM];
    const int col = threadIdx.x & (DIM - 1);
    const int rg  = threadIdx.x >> 7;        // 0 or 1
    const long row0 = (long)blockIdx.x * 128 + rg;
    float s = 0.f, q = 0.f;
#pragma unroll 4
    for (int i = 0; i < 64; ++i) {
        long r = row0 + 2 * i;
        if (r < N_NODES) {
            float v = out[r * DIM + col];
            s += v; q += v * v;
        }
    }
    s_sum[rg][col] = s;
    s_sq[rg][col]  = q;
    __syncthreads();
    if (rg == 0) {
        atomicAdd(&colsum[col], s_sum[0][col] + s_sum[1][col]);
        atomicAdd(&colsq[col],  s_sq[0][col]  + s_sq[1][col]);
    }
}

// ---------------------------------------------------------------------------
// Kernel 5: BatchNorm (batch stats, biased var) + ReLU, in place on out
// ---------------------------------------------------------------------------
__global__ __launch_bounds__(256) void bn_relu(
        float* __restrict__ out, const float* __restrict__ colsum,
        const float* __restrict__ colsq, const float* __restrict__ gamma,
        const float* __restrict__ beta) {
    const float invN = 1.0f / (float)N_NODES;
    long i = (long)blockIdx.x * blockDim.x + threadIdx.x;
    const long n = (long)N_NODES * DIM;
    const long stride = (long)gridDim.x * blockDim.x;
    for (; i < n; i += stride) {
        const int col = (int)(i & (DIM - 1));
        const float mean = colsum[col] * invN;
        const float var  = colsq[col] * invN - mean * mean;
        const float scl  = rsqrtf(var + BN_EPS) * gamma[col];
        float v = (out[i] - mean) * scl + beta[col];
        out[i] = v > 0.f ? v : 0.f;
    }
}

// ---------------------------------------------------------------------------
// Kernel 6: rel_out = relu(rel_emb @ W_rel^T), 500x128. WMMA with clamped
// tail rows (keeps EXEC all-ones + in-bounds loads) and masked stores.
// ---------------------------------------------------------------------------
__global__ __launch_bounds__(256) void rel_gemm(
        const float* __restrict__ rel, const float* __restrict__ Wrel,
        float* __restrict__ rout) {
    const int lane = threadIdx.x & 31;
    const int wave = threadIdx.x >> 5;
    const int m0 = blockIdx.x * 16;
    const int n0 = wave * 16;
    const int ml = lane & 15;
    const int kh = (lane >> 4) * 2;

    int arow = m0 + ml;
    if (arow >= N_RELS) arow = N_RELS - 1;   // clamp (duplicate last row)

    v8f c = {};
    c = wmma_k128(c, rel + (long)arow * DIM + kh,
                     Wrel + (long)(n0 + ml) * DIM + kh);

    const int rbase = m0 + (lane >> 4) * 8;
#pragma unroll
    for (int r = 0; r < 8; ++r) {
        const int row = rbase + r;
        if (row < N_RELS) {
            float v = c[r];
            rout[(long)row * DIM + n0 + ml] = v > 0.f ? v : 0.f;
        }
    }
}

// ---------------------------------------------------------------------------
extern "C" void kernel_launch(void* const* d_in, const int* in_sizes, int n_in,
                              void* d_out, int out_size, void* d_ws, size_t ws_size,
                              hipStream_t stream) {
    const float* ent   = (const float*)d_in[0];
    const float* rel   = (const float*)d_in[1];
    const int*   eidx  = (const int*)d_in[2];
    const int*   etype = (const int*)d_in[3];
    const int*   edir  = (const int*)d_in[4];
    const float* enorm = (const float*)d_in[5];
    const float* Wout  = (const float*)d_in[6];
    const float* Win   = (const float*)d_in[7];
    const float* Wloop = (const float*)d_in[8];
    const float* Wrel  = (const float*)d_in[9];
    const float* gamma = (const float*)d_in[10];
    const float* beta  = (const float*)d_in[11];

    float* out  = (float*)d_out;                     // [N_NODES, DIM]
    float* rout = out + (long)N_NODES * DIM;         // [N_RELS, DIM]

    float* ws     = (float*)d_ws;
    float* acc    = ws;                               // 3 * N_NODES * DIM
    float* colsum = ws + 3L * N_NODES * DIM;          // DIM
    float* colsq  = colsum + DIM;                     // DIM

    const long n_zero = 3L * N_NODES * DIM + 2 * DIM; // multiple of 4
    zero_ws4<<<2048, 256, 0, stream>>>((float4*)ws, n_zero / 4);

    scatter_edges<<<(N_EDGES + 7) / 8, 256, 0, stream>>>(
        ent, rel, eidx, etype, edir, enorm, acc);

    gemm_out<<<N_NODES / 16, 256, 0, stream>>>(acc, Wout, Win, Wloop, out);

    col_stats<<<(N_NODES + 127) / 128, 256, 0, stream>>>(out, colsum, colsq);

    bn_relu<<<4096, 256, 0, stream>>>(out, colsum, colsq, gamma, beta);

    rel_gemm<<<(N_RELS + 15) / 16, 256, 0, stream>>>(rel, Wrel, rout);
}